// LocSE_59691455480199
// MI455X (gfx1250) — compile-verified
//
#include <hip/hip_runtime.h>
#include <math.h>

// LocSE: out(B,N,17,64) = concat( relu(rppe(17x53) @ W(53x32) + b), feats[nidx](17x32) )
// B=4, N=32768 (2^15), K=16 -> Kp1=17, CH=53 (pad to 56), UNITS=32.
// Memory-bound (~600MB HBM): stream output, gathers hit L2 (pc=1MB, feats=16MB << 192MB L2).

#define KP1     17
#define UNITS   32
#define CH      53
#define NSTEPS  14          // ceil(56/4) K-steps of 16x16x4 f32 WMMA
#define ROWS    16          // M rows per WMMA tile
#define RSTR    58          // LDS row stride in floats (58 mod 64 -> conflict-free, even -> 8B align)
#define WAVES   8           // waves per block (256 threads, wave32)
#define TPW     8           // tiles per wave (amortize W/bias preload)
#define LOGN    15          // N = 32768

typedef __attribute__((ext_vector_type(2))) float v2f;
typedef __attribute__((ext_vector_type(4))) float v4f;
typedef __attribute__((ext_vector_type(8))) float v8f;

__global__ __launch_bounds__(WAVES * 32)
void locse_wmma_kernel(const float* __restrict__ pc,
                       const float* __restrict__ feats,
                       const float* __restrict__ W,
                       const float* __restrict__ bias,
                       const int*   __restrict__ nidx,
                       float*       __restrict__ out,
                       int ntiles, int npts /* B*N */)
{
    __shared__ float  rppe[WAVES][ROWS * RSTR];      // 16 x 56 A-tile per wave (padded stride)
    __shared__ float4 nbs[WAVES][2][KP1];            // {x, y, norm, 0} neighborhoods of <=2 points

    const int lane = threadIdx.x & 31;
    // wave index is wave-uniform: force it into an SGPR so all control flow derived
    // from it lowers to scalar branches instead of exec-mask manipulation.
    const int w    = __builtin_amdgcn_readfirstlane(threadIdx.x >> 5);
    const int ncol = lane & 15;          // N column within 16-wide tile
    const int hi   = lane >> 4;          // half-wave select
    const int rtot = npts * KP1;

    // ---- preload B fragments: W is (53 x 32) row-major, K padded to 56 with zeros ----
    // B layout (16x16x4): VGPR0 = B[k0 + 2*hi][n], VGPR1 = B[k0 + 2*hi + 1][n], n = lane&15
    v2f Bf0[NSTEPS], Bf1[NSTEPS];
    {
        const int kb = hi * 2;
#pragma unroll
        for (int s = 0; s < NSTEPS; ++s) {
            const int k0 = 4 * s + kb, k1 = k0 + 1;
            Bf0[s].x = (k0 < CH) ? W[k0 * UNITS + ncol]      : 0.f;
            Bf0[s].y = (k1 < CH) ? W[k1 * UNITS + ncol]      : 0.f;
            Bf1[s].x = (k0 < CH) ? W[k0 * UNITS + 16 + ncol] : 0.f;
            Bf1[s].y = (k1 < CH) ? W[k1 * UNITS + 16 + ncol] : 0.f;
        }
    }
    const float bias0 = bias[ncol];
    const float bias1 = bias[16 + ncol];

    const int wave_global = blockIdx.x * WAVES + w;

    for (int t = 0; t < TPW; ++t) {
        // all wave-uniform scalars (SGPR): tile, r0, full, p0, p1
        const int tile = __builtin_amdgcn_readfirstlane(wave_global * TPW + t);
        if (tile >= ntiles) break;                    // scalar branch
        const int r0   = tile * ROWS;
        const bool full = (r0 + ROWS <= rtot);        // always true when rtot % 16 == 0
        const int p0   = r0 / KP1;                    // first point in tile
        const int p1   = (r0 + ROWS - 1) / KP1;       // last point (p0 or p0+1)

        // ---- gather neighborhood(s) {pos, norm} into per-wave LDS table ----
        if (lane < KP1) {
            const int idx  = nidx[p0 * KP1 + lane];
            const int base = ((p0 >> LOGN) << LOGN) + idx;    // b*N + idx
            const float px = pc[base * 2 + 0];
            const float py = pc[base * 2 + 1];
            nbs[w][0][lane] = make_float4(px, py, sqrtf(px * px + py * py), 0.f);
        }
        if (p1 != p0 && p1 < npts) {                  // scalar condition
            if (lane < KP1) {
                const int idx  = nidx[p1 * KP1 + lane];
                const int base = ((p1 >> LOGN) << LOGN) + idx;
                const float px = pc[base * 2 + 0];
                const float py = pc[base * 2 + 1];
                nbs[w][1][lane] = make_float4(px, py, sqrtf(px * px + py * py), 0.f);
            }
        }
        // Same-wave DS ops are in-order on CDNA5; per-wave LDS region -> no block barrier.

        // ---- build 16 x 56 rppe tile in LDS; lane covers half a row ----
        const int m = ncol;                                  // tile row
        const int r = r0 + m;                                // global row (point, slot)
        const bool rvalid = full | (r < rtot);
        const int p  = rvalid ? (r / KP1) : p0;
        const int a  = rvalid ? (r - p * KP1) : 0;
        const int pi = p - p0;                               // 0 or 1
        float* rowp = &rppe[w][m * RSTR];
        const float4 pa4 = nbs[w][pi][a];                    // own point position
        if (hi == 0) {                                       // columns 0..27
            rowp[0] = pa4.x;
            rowp[1] = pa4.y;
#pragma unroll
            for (int c = 0; c < 8; ++c) {
                const float4 nc = nbs[w][pi][c];
                rowp[2 + 3 * c + 0] = nc.x - pa4.x;
                rowp[2 + 3 * c + 1] = nc.y - pa4.y;
                rowp[2 + 3 * c + 2] = nc.z;
            }
            const float4 n8 = nbs[w][pi][8];
            rowp[26] = n8.x - pa4.x;
            rowp[27] = n8.y - pa4.y;
        } else {                                             // columns 28..55
            const float4 n8 = nbs[w][pi][8];
            rowp[28] = n8.z;
#pragma unroll
            for (int c = 9; c < KP1; ++c) {
                const float4 nc = nbs[w][pi][c];
                rowp[2 + 3 * c + 0] = nc.x - pa4.x;
                rowp[2 + 3 * c + 1] = nc.y - pa4.y;
                rowp[2 + 3 * c + 2] = nc.z;
            }
            rowp[53] = 0.f; rowp[54] = 0.f; rowp[55] = 0.f;  // K padding
        }

        // ---- (16x56) @ (56x32) via 14 x 2 V_WMMA_F32_16X16X4_F32, bias-seeded ----
        v8f acc0, acc1;
#pragma unroll
        for (int i = 0; i < 8; ++i) { acc0[i] = bias0; acc1[i] = bias1; }
        // A layout: VGPR0 = A[m][k0 + 2*hi], VGPR1 = A[m][k0 + 2*hi + 1]
        const float* arow = &rppe[w][m * RSTR + hi * 2];
#pragma unroll
        for (int s = 0; s < NSTEPS; ++s) {
            const v2f av = *(const v2f*)(arow + 4 * s);      // ds_load_2addr_b64, conflict-free
            acc0 = __builtin_amdgcn_wmma_f32_16x16x4_f32(
                       false, av, false, Bf0[s], (short)0, acc0, false, false);
            acc1 = __builtin_amdgcn_wmma_f32_16x16x4_f32(
                       false, av, false, Bf1[s], (short)0, acc1, false, false);
        }

        // ---- ReLU + streaming (nontemporal) stores ----
        if (full) {
            // fast path (scalar-branch selected): one base address, immediate offsets
            float* o0 = out + (size_t)(r0 + hi * 8) * 64 + ncol;   // D: VGPR i -> M = i + 8*hi
#pragma unroll
            for (int i = 0; i < 8; ++i) {
                __builtin_nontemporal_store(fmaxf(acc0[i], 0.f), o0 + (size_t)i * 64);
                __builtin_nontemporal_store(fmaxf(acc1[i], 0.f), o0 + (size_t)i * 64 + 16);
            }
            // feats gather -> out[..., 32:64] (16 f32 per lane, b128 streaming stores)
            const int idx   = nidx[r];
            const int gbase = (((p >> LOGN) << LOGN) + idx) * UNITS + hi * 16;
            float* o1 = out + (size_t)r * 64 + 32 + hi * 16;
#pragma unroll
            for (int j = 0; j < 4; ++j) {
                const v4f f = *(const v4f*)(feats + gbase + 4 * j);
                __builtin_nontemporal_store(f, (v4f*)(o1 + 4 * j));
            }
        } else {
            // tail path (never taken for the harness shape): per-row guards
#pragma unroll
            for (int i = 0; i < 8; ++i) {
                const int rr = r0 + i + hi * 8;
                if (rr < rtot) {
                    const size_t ob = (size_t)rr * 64;
                    __builtin_nontemporal_store(fmaxf(acc0[i], 0.f), out + ob + ncol);
                    __builtin_nontemporal_store(fmaxf(acc1[i], 0.f), out + ob + 16 + ncol);
                }
            }
            if (r < rtot) {
                const int idx   = nidx[r];
                const int gbase = (((p >> LOGN) << LOGN) + idx) * UNITS + hi * 16;
                const size_t ob = (size_t)r * 64 + 32 + hi * 16;
#pragma unroll
                for (int j = 0; j < 4; ++j) {
                    const v4f f = *(const v4f*)(feats + gbase + 4 * j);
                    __builtin_nontemporal_store(f, (v4f*)(out + ob + 4 * j));
                }
            }
        }
    }
}

extern "C" void kernel_launch(void* const* d_in, const int* in_sizes, int n_in,
                              void* d_out, int out_size, void* d_ws, size_t ws_size,
                              hipStream_t stream) {
    const float* pc    = (const float*)d_in[0];
    const float* feats = (const float*)d_in[1];
    const float* W     = (const float*)d_in[2];
    const float* bias  = (const float*)d_in[3];
    const int*   nidx  = (const int*)d_in[4];
    float*       out   = (float*)d_out;

    const int npts   = in_sizes[0] / 2;                       // B*N = 131072
    const int rtot   = npts * KP1;                            // 2,228,224 rows
    const int ntiles = (rtot + ROWS - 1) / ROWS;              // 139,264 (exact)
    const int waves  = (ntiles + TPW - 1) / TPW;
    const int blocks = (waves + WAVES - 1) / WAVES;           // 2,176 blocks

    locse_wmma_kernel<<<dim3(blocks), dim3(WAVES * 32), 0, stream>>>(
        pc, feats, W, bias, nidx, out, ntiles, npts);
}